// Sinkhorn_32427003085120
// MI455X (gfx1250) — compile-verified
//
#include <hip/hip_runtime.h>
#include <math.h>

#define NT 2048
#define NR 128
#define NB 8
#define NIT 32
#define EPS_F 2.5e-9f           // blur^2, blur = 5e-5
#define INV_EPS_F 4.0e8f        // 1/EPS
#define LOGC (-7.6246189861593985f)   // -log(2048) = loga = logb
#define OUT_SCALE (EPS_F / 2048.0f)

typedef float v2f __attribute__((ext_vector_type(2)));
typedef float v8f __attribute__((ext_vector_type(8)));

struct LseTask  { const float* mat; const float* vec; float* out; int sym; };
struct LseArgs  { LseTask t[6]; };
struct GemmTask { const float* X; const float* Y; const float* hx; const float* hy; float* out; };
struct GemmArgs { GemmTask t[8]; };

// --- column sums of x^2 over batch 0 (reference quirk: wx from sample 0 only) ---
__global__ __launch_bounds__(128) void wsum_k(const float* __restrict__ x,
                                              const float* __restrict__ y,
                                              float* __restrict__ invw) {
    const float* src = blockIdx.x ? y : x;   // batch 0 region of each input
    int r = threadIdx.x;                     // 0..127
    float s = 0.0f;
#pragma unroll 8
    for (int n = 0; n < NT; ++n) {
        float v = src[(size_t)n * NR + r];
        s += v * v;
    }
    invw[blockIdx.x * 128 + r] = 1.0f / s;
}

// --- xb = x^2 * invw (broadcast), plus hx[i] = 0.5*sum_r xb^2 ---
__global__ __launch_bounds__(256) void norm_k(const float* __restrict__ x,
                                              const float* __restrict__ y,
                                              const float* __restrict__ invw,
                                              float* __restrict__ xb, float* __restrict__ yb,
                                              float* __restrict__ hx, float* __restrict__ hy) {
    int wave = threadIdx.x >> 5, lane = threadIdx.x & 31;
    int row  = blockIdx.x * 8 + wave;               // 0 .. NB*NT-1
    const float* src = blockIdx.y ? y : x;
    const float* iw  = invw + (blockIdx.y ? 128 : 0);
    float* dst = blockIdx.y ? yb : xb;
    float* h   = blockIdx.y ? hy : hx;

    size_t base = (size_t)row * NR + lane * 4;
    float4 a = *(const float4*)(src + base);
    float4 w = *(const float4*)(iw + lane * 4);
    float4 o;
    o.x = a.x * a.x * w.x; o.y = a.y * a.y * w.y;
    o.z = a.z * a.z * w.z; o.w = a.w * a.w * w.w;
    *(float4*)(dst + base) = o;

    float s = o.x * o.x + o.y * o.y + o.z * o.z + o.w * o.w;
#pragma unroll
    for (int off = 16; off; off >>= 1) s += __shfl_xor(s, off, 32);
    if (lane == 0) h[row] = 0.5f * s;
}

__global__ __launch_bounds__(256) void zero_k(float* __restrict__ p) {
    p[(size_t)blockIdx.x * 256 + threadIdx.x] = 0.0f;
}

// --- cost matrix via f32 WMMA: out[i,j] = (hx[i] + hy[j] - dot(X_i, Y_j)) / EPS ---
__global__ __launch_bounds__(256) void gemm_k(GemmArgs ga) {
    GemmTask t = ga.t[blockIdx.y];
    int wave = threadIdx.x >> 5, lane = threadIdx.x & 31;
    int tile = blockIdx.x * 8 + wave;       // 0..16383
    int br = tile >> 7, bc = tile & 127;

    int r    = lane & 15;
    int koff = (lane >> 4) * 2;             // A/B 32-bit fragment layout
    const float* xp = t.X + (size_t)(br * 16 + r) * NR + koff;
    const float* yp = t.Y + (size_t)(bc * 16 + r) * NR + koff;

    v8f acc = {0.f, 0.f, 0.f, 0.f, 0.f, 0.f, 0.f, 0.f};
#pragma unroll
    for (int k = 0; k < NR; k += 4) {
        v2f a = *(const v2f*)(xp + k);
        v2f b = *(const v2f*)(yp + k);
        acc = __builtin_amdgcn_wmma_f32_16x16x4_f32(
            false, a, false, b, (short)0, acc, false, false);
    }

    int j     = bc * 16 + (lane & 15);
    int ibase = br * 16 + (lane >> 4) * 8;  // C layout: VGPR v -> M = v (+8 for hi lanes)
    float hyj = t.hy[j];
#pragma unroll
    for (int v = 0; v < 8; ++v) {
        int i = ibase + v;
        float val = (t.hx[i] + hyj - acc[v]) * INV_EPS_F;
        t.out[(size_t)i * NT + j] = val;
    }
}

// --- row-wise LSE update; one wave per row, whole row in registers ---
__global__ __launch_bounds__(256) void lse_k(LseArgs la) {
    LseTask t = la.t[blockIdx.y];
    int wave = threadIdx.x >> 5, lane = threadIdx.x & 31;
    int row  = blockIdx.x * 8 + wave;
    const float* mrow = t.mat + (size_t)row * NT;

    float4 d[16];
#pragma unroll
    for (int u = 0; u < 16; ++u) {
        int j = u * 128 + lane * 4;
        float4 m = *(const float4*)(mrow + j);
        float4 v = *(const float4*)(t.vec + j);
        d[u].x = v.x - m.x; d[u].y = v.y - m.y;
        d[u].z = v.z - m.z; d[u].w = v.w - m.w;
    }
    float mx = -3.402823466e38f;
#pragma unroll
    for (int u = 0; u < 16; ++u)
        mx = fmaxf(mx, fmaxf(fmaxf(d[u].x, d[u].y), fmaxf(d[u].z, d[u].w)));
#pragma unroll
    for (int off = 16; off; off >>= 1) mx = fmaxf(mx, __shfl_xor(mx, off, 32));

    float s = 0.0f;
#pragma unroll
    for (int u = 0; u < 16; ++u)
        s += __expf(d[u].x - mx) + __expf(d[u].y - mx) +
             __expf(d[u].z - mx) + __expf(d[u].w - mx);
#pragma unroll
    for (int off = 16; off; off >>= 1) s += __shfl_xor(s, off, 32);

    if (lane == 0) {
        float L = LOGC + mx + __logf(s);
        t.out[row] = t.sym ? 0.5f * (t.vec[row] - L) : -L;
    }
}

// --- sum_b EPS*(mean F + mean G - mean Px - mean Py) ---
__global__ __launch_bounds__(256) void final_k(const float* __restrict__ pot,
                                               float* __restrict__ out) {
    __shared__ float red[256];
    float acc = 0.0f;
    for (int b = 0; b < NB; ++b) {
        const float* base = pot + (size_t)b * 6 * NT;
        for (int i = threadIdx.x; i < NT; i += 256)
            acc += base[i] + base[NT + i] - base[2 * NT + i] - base[4 * NT + i];
    }
    red[threadIdx.x] = acc;
    __syncthreads();
    for (int s = 128; s > 0; s >>= 1) {
        if (threadIdx.x < s) red[threadIdx.x] += red[threadIdx.x + s];
        __syncthreads();
    }
    if (threadIdx.x == 0) out[0] = red[0] * OUT_SCALE;
}

extern "C" void kernel_launch(void* const* d_in, const int* in_sizes, int n_in,
                              void* d_out, int out_size, void* d_ws, size_t ws_size,
                              hipStream_t stream) {
    (void)in_sizes; (void)n_in; (void)out_size; (void)ws_size;
    const float* x = (const float*)d_in[0];
    const float* y = (const float*)d_in[1];
    float* out = (float*)d_out;
    float* ws  = (float*)d_ws;

    size_t o = 0;
    float* invw = ws + o; o += 256;
    float* xb   = ws + o; o += (size_t)NB * NT * NR;
    float* yb   = ws + o; o += (size_t)NB * NT * NR;
    float* hx   = ws + o; o += (size_t)NB * NT;
    float* hy   = ws + o; o += (size_t)NB * NT;
    float* pot  = ws + o; o += (size_t)NB * 6 * NT;   // per batch: F,G,Px0,Px1,Py0,Py1
    float* mats = ws + o;                              // 2 slots * 4 * NT*NT
    const size_t MSZ = (size_t)NT * NT;

    wsum_k<<<dim3(2), 128, 0, stream>>>(x, y, invw);
    norm_k<<<dim3(NB * NT / 8, 2), 256, 0, stream>>>(x, y, invw, xb, yb, hx, hy);
    zero_k<<<(NB * 6 * NT) / 256, 256, 0, stream>>>(pot);

    for (int pair = 0; pair < 4; ++pair) {
        float* slot[2];
        GemmArgs ga;
        for (int s = 0; s < 2; ++s) {
            int b = pair * 2 + s;
            slot[s] = mats + (size_t)s * 4 * MSZ;
            const float* X   = xb + (size_t)b * NT * NR;
            const float* Y   = yb + (size_t)b * NT * NR;
            const float* hxb = hx + (size_t)b * NT;
            const float* hyb = hy + (size_t)b * NT;
            ga.t[s * 4 + 0] = {X, Y, hxb, hyb, slot[s] + 0 * MSZ};  // Ce(x,y)
            ga.t[s * 4 + 1] = {Y, X, hyb, hxb, slot[s] + 1 * MSZ};  // Ce(x,y)^T
            ga.t[s * 4 + 2] = {X, X, hxb, hxb, slot[s] + 2 * MSZ};  // Ce(x,x)
            ga.t[s * 4 + 3] = {Y, Y, hyb, hyb, slot[s] + 3 * MSZ};  // Ce(y,y)
        }
        gemm_k<<<dim3(2048, 8), 256, 0, stream>>>(ga);

        for (int it = 0; it < NIT; ++it) {
            int r = it & 1;
            LseArgs A{}, B{};
            for (int s = 0; s < 2; ++s) {
                int b = pair * 2 + s;
                float* F   = pot + (size_t)b * 6 * NT;
                float* G   = F + NT;
                float* Pxr = F + (size_t)(2 + r) * NT;
                float* Pxw = F + (size_t)(2 + (r ^ 1)) * NT;
                float* Pyr = F + (size_t)(4 + r) * NT;
                float* Pyw = F + (size_t)(4 + (r ^ 1)) * NT;
                A.t[s * 3 + 0] = {slot[s] + 0 * MSZ, G,   F,   0};  // f-update
                A.t[s * 3 + 1] = {slot[s] + 2 * MSZ, Pxr, Pxw, 1};  // px-update
                A.t[s * 3 + 2] = {slot[s] + 3 * MSZ, Pyr, Pyw, 1};  // py-update
                B.t[s]         = {slot[s] + 1 * MSZ, F,   G,   0};  // g-update (uses new F)
            }
            lse_k<<<dim3(256, 6), 256, 0, stream>>>(A);
            lse_k<<<dim3(256, 2), 256, 0, stream>>>(B);
        }
    }
    final_k<<<1, 256, 0, stream>>>(pot, out);
}